// Model_55147380081070
// MI455X (gfx1250) — compile-verified
//
#include <hip/hip_runtime.h>
#include <stdint.h>
#include <math.h>

#define B_SZ   8192
#define NE_SZ  14952
#define NR_SZ  1345
#define DIM    256
#define NC_SZ  10
#define PC_SZ  10
#define L_SZ   3

#define WAVES_PER_BLOCK 8
#define BLOCK_THREADS   256
#define NBLK            (B_SZ / WAVES_PER_BLOCK)   // 1024 blocks

#define WAVE_BUF_BYTES  3072                        // one tile buffer (max 3 rows x 1KB)
#define WAVE_LDS_BYTES  (2 * WAVE_BUF_BYTES)        // double buffered

// s_wait_asynccnt <= n : oldest async copies are complete (in-order completion)
#define S_WAIT_ASYNC(n) asm volatile("s_wait_asynccnt " #n ::: "memory")

// Copy one 1KB embedding row (256 fp32) global -> LDS with the CDNA5 async engine.
// Each of the 32 lanes moves 16B at byte offset 16*lane; the +512 instruction
// offset applies to BOTH the LDS and global address per the ISA pseudocode.
__device__ __forceinline__ void async_row_1k(uint32_t lds_addr, const float* grow, int lane) {
  const char* g = (const char*)grow + (lane << 4);
  uint32_t    l = lds_addr + (lane << 4);
  asm volatile(
      "global_load_async_to_lds_b128 %0, %1, off\n\t"
      "global_load_async_to_lds_b128 %0, %1, off offset:512"
      :: "v"(l), "v"(g)
      : "memory");
}

__device__ __forceinline__ float wave_sum32(float v) {
  #pragma unroll
  for (int m = 16; m >= 1; m >>= 1) v += __shfl_xor(v, m, 32);
  return v;  // fixed tree -> bitwise deterministic
}

__device__ __forceinline__ float log_sigmoid(float x) {
  // numerically stable: ls(x) = min(x,0) - log1p(exp(-|x|))
  return fminf(x, 0.0f) - log1pf(expf(-fabsf(x)));
}

__device__ __forceinline__ void load_row8(const float* table, int row, int lane, float v[8]) {
  const float* p = table + (size_t)row * DIM + (lane << 2);
  float4 lo = *(const float4*)(p);
  float4 hi = *(const float4*)(p + 128);
  v[0] = lo.x; v[1] = lo.y; v[2] = lo.z; v[3] = lo.w;
  v[4] = hi.x; v[5] = hi.y; v[6] = hi.z; v[7] = hi.w;
}

__device__ __forceinline__ void lds_row8(const char* buf, int lane, float v[8]) {
  float4 lo = *(const float4*)(buf + (lane << 4));
  float4 hi = *(const float4*)(buf + 512 + (lane << 4));
  v[0] = lo.x; v[1] = lo.y; v[2] = lo.z; v[3] = lo.w;
  v[4] = hi.x; v[5] = hi.y; v[6] = hi.z; v[7] = hi.w;
}

__global__ void __launch_bounds__(BLOCK_THREADS)
ptranse_main(const int* __restrict__ h, const int* __restrict__ r, const int* __restrict__ t,
             const int* __restrict__ hneg, const int* __restrict__ rneg, const int* __restrict__ tneg,
             const int* __restrict__ nctx,   // (B, NC, 2) int32
             const int* __restrict__ pctx,   // (B, PC, L) int32
             const float* __restrict__ ent,  // (NE, 256)
             const float* __restrict__ rel,  // (NR, 256)
             float* __restrict__ o_gnp, float* __restrict__ o_gnn, float* __restrict__ o_pl)
{
  __shared__ __align__(16) char smem[WAVES_PER_BLOCK * WAVE_LDS_BYTES];
  __shared__ float partial[WAVES_PER_BLOCK][3];

  const int lane = threadIdx.x & 31;
  const int wave = threadIdx.x >> 5;
  const int b    = blockIdx.x * WAVES_PER_BLOCK + wave;

  char* swave = smem + wave * WAVE_LDS_BYTES;
  const uint32_t lbase = (uint32_t)(uintptr_t)swave;  // low 32 bits of flat addr == LDS offset

  // ---- base rows: lane owns d in {4l..4l+3, 128+4l..128+4l+3} ----
  float eh[8], er[8], et[8], ehn[8], ern[8], etn[8];
  load_row8(ent, __builtin_amdgcn_readfirstlane(h[b]),    lane, eh);
  load_row8(rel, __builtin_amdgcn_readfirstlane(r[b]),    lane, er);
  load_row8(ent, __builtin_amdgcn_readfirstlane(t[b]),    lane, et);
  load_row8(ent, __builtin_amdgcn_readfirstlane(hneg[b]), lane, ehn);
  load_row8(ent, __builtin_amdgcn_readfirstlane(rneg[b]), lane, ern);  // reference: entity_emb[r_neg]
  load_row8(ent, __builtin_amdgcn_readfirstlane(tneg[b]), lane, etn);

  // ---- neighbor indices (wave-uniform -> SGPRs) ----
  int rIdx[NC_SZ], tIdx[NC_SZ];
  #pragma unroll
  for (int i = 0; i < NC_SZ; ++i) {
    rIdx[i] = __builtin_amdgcn_readfirstlane(nctx[((size_t)b * NC_SZ + i) * 2 + 0]);
    tIdx[i] = __builtin_amdgcn_readfirstlane(nctx[((size_t)b * NC_SZ + i) * 2 + 1]);
  }

  // kick off first neighbor tile: relation row @+0, entity row @+1024
  async_row_1k(lbase + 0,    rel + (size_t)rIdx[0] * DIM, lane);
  async_row_1k(lbase + 1024, ent + (size_t)tIdx[0] * DIM, lane);

  float aN[NC_SZ], nP[NC_SZ], nN[NC_SZ];
  #pragma unroll
  for (int nc = 0; nc < NC_SZ; ++nc) {
    if (nc + 1 < NC_SZ) {
      uint32_t nb = lbase + ((nc + 1) & 1) * WAVE_BUF_BYTES;
      async_row_1k(nb,        rel + (size_t)rIdx[nc + 1] * DIM, lane);
      async_row_1k(nb + 1024, ent + (size_t)tIdx[nc + 1] * DIM, lane);
      S_WAIT_ASYNC(4);   // 4 newer copies may stay in flight; current tile done
    } else {
      S_WAIT_ASYNC(0);
    }
    const char* buf = swave + (nc & 1) * WAVE_BUF_BYTES;
    float rr[8], tt[8];
    lds_row8(buf,        lane, rr);
    lds_row8(buf + 1024, lane, tt);
    float s1 = 0.f, s2 = 0.f, s3 = 0.f;
    #pragma unroll
    for (int j = 0; j < 8; ++j) {
      float tmp = rr[j] - tt[j];            // neighbor_tmp
      float d1 = er[j] - et[j] - tmp;       // -tmp + er - et
      float d2 = eh[j]  + tmp;
      float d3 = ehn[j] + tmp;
      s1 = fmaf(d1, d1, s1);
      s2 = fmaf(d2, d2, s2);
      s3 = fmaf(d3, d3, s3);
    }
    aN[nc] = sqrtf(wave_sum32(s1));
    nP[nc] = sqrtf(wave_sum32(s2));
    nN[nc] = sqrtf(wave_sum32(s3));
  }

  // ---- path indices ----
  int pIdx[PC_SZ][L_SZ]; float pSgn[PC_SZ][L_SZ];
  #pragma unroll
  for (int pc = 0; pc < PC_SZ; ++pc) {
    #pragma unroll
    for (int l = 0; l < L_SZ; ++l) {
      int v = __builtin_amdgcn_readfirstlane(pctx[((size_t)b * PC_SZ + pc) * L_SZ + l]);
      pSgn[pc][l] = (float)((v > 0) - (v < 0));   // sign(0) = 0 like jnp.sign
      pIdx[pc][l] = v < 0 ? -v : v;
    }
  }

  // kick off first path tile (3 relation rows) before doing the neighbor softmax
  #pragma unroll
  for (int l = 0; l < L_SZ; ++l)
    async_row_1k(lbase + l * 1024, rel + (size_t)pIdx[0][l] * DIM, lane);

  // neighbor softmax over -a, overlapped with the async path tile
  float mn = aN[0];
  #pragma unroll
  for (int nc = 1; nc < NC_SZ; ++nc) mn = fminf(mn, aN[nc]);
  float den = 0.f, wN[NC_SZ];
  #pragma unroll
  for (int nc = 0; nc < NC_SZ; ++nc) { wN[nc] = expf(mn - aN[nc]); den += wN[nc]; }
  float inv = 1.0f / den;
  float gnp = 0.f, gnn = 0.f;
  #pragma unroll
  for (int nc = 0; nc < NC_SZ; ++nc) {
    float al = wN[nc] * inv;
    gnp += al * nP[nc];
    gnn += al * nN[nc];
  }

  // ---- path loop ----
  float sB[PC_SZ], sR[PC_SZ], sBn[PC_SZ], sRn[PC_SZ];
  #pragma unroll
  for (int pc = 0; pc < PC_SZ; ++pc) {
    if (pc + 1 < PC_SZ) {
      uint32_t nb = lbase + ((pc + 1) & 1) * WAVE_BUF_BYTES;
      #pragma unroll
      for (int l = 0; l < L_SZ; ++l)
        async_row_1k(nb + l * 1024, rel + (size_t)pIdx[pc + 1][l] * DIM, lane);
      S_WAIT_ASYNC(6);   // 6 newer copies in flight; current 3 rows done
    } else {
      S_WAIT_ASYNC(0);
    }
    const char* buf = swave + (pc & 1) * WAVE_BUF_BYTES;
    float r0[8], r1[8], r2[8];
    lds_row8(buf,        lane, r0);
    lds_row8(buf + 1024, lane, r1);
    lds_row8(buf + 2048, lane, r2);
    float s1 = 0.f, s2 = 0.f, s3 = 0.f, s4 = 0.f;
    #pragma unroll
    for (int j = 0; j < 8; ++j) {
      float ep = pSgn[pc][0] * r0[j] + pSgn[pc][1] * r1[j] + pSgn[pc][2] * r2[j];
      float d1 = eh[j] + ep - et[j];    // b  == ||path_tmp - et||
      float d2 = ep - er[j];
      float d3 = eh[j] + ep - etn[j];
      float d4 = ep - ern[j];
      s1 = fmaf(d1, d1, s1);
      s2 = fmaf(d2, d2, s2);
      s3 = fmaf(d3, d3, s3);
      s4 = fmaf(d4, d4, s4);
    }
    sB[pc]  = sqrtf(wave_sum32(s1));
    sR[pc]  = sqrtf(wave_sum32(s2));
    sBn[pc] = sqrtf(wave_sum32(s3));
    sRn[pc] = sqrtf(wave_sum32(s4));
  }

  // path softmax over -b
  float mb = sB[0];
  #pragma unroll
  for (int pc = 1; pc < PC_SZ; ++pc) mb = fminf(mb, sB[pc]);
  den = 0.f;
  float wP[PC_SZ];
  #pragma unroll
  for (int pc = 0; pc < PC_SZ; ++pc) { wP[pc] = expf(mb - sB[pc]); den += wP[pc]; }
  inv = 1.0f / den;
  float gpp = 0.f, gpn = 0.f;
  #pragma unroll
  for (int pc = 0; pc < PC_SZ; ++pc) {
    float be = wP[pc] * inv;
    gpp += be * (sB[pc]  + sR[pc]);
    gpn += be * (sBn[pc] + sRn[pc]);
  }
  float pl = -log_sigmoid(gpp) - log_sigmoid(-gpn);

  // deterministic intra-block combine (fixed order, no atomics)
  if (lane == 0) { partial[wave][0] = gnp; partial[wave][1] = gnn; partial[wave][2] = pl; }
  __syncthreads();
  if (threadIdx.x == 0) {
    float A = 0.f, Bc = 0.f, C = 0.f;
    #pragma unroll
    for (int w = 0; w < WAVES_PER_BLOCK; ++w) {
      A  += partial[w][0];
      Bc += partial[w][1];
      C  += partial[w][2];
    }
    o_gnp[blockIdx.x] = A;
    o_gnn[blockIdx.x] = Bc;
    o_pl[blockIdx.x]  = C;
  }
}

__global__ void __launch_bounds__(256)
ptranse_finish(const float* __restrict__ gnp, const float* __restrict__ gnn,
               const float* __restrict__ pl, float* __restrict__ out)
{
  __shared__ float s0[256], s1[256], s2[256];
  const int tid = threadIdx.x;
  float a = 0.f, b = 0.f, c = 0.f;
  for (int i = tid; i < NBLK; i += 256) { a += gnp[i]; b += gnn[i]; c += pl[i]; }
  s0[tid] = a; s1[tid] = b; s2[tid] = c;
  __syncthreads();
  for (int off = 128; off >= 1; off >>= 1) {
    if (tid < off) { s0[tid] += s0[tid + off]; s1[tid] += s1[tid + off]; s2[tid] += s2[tid + off]; }
    __syncthreads();
  }
  if (tid == 0) {
    // loss = B*(-ls(g_n_pos) - ls(-g_n_neg)) + sum_b(-ls(g_p_pos) - ls(-g_p_neg))
    out[0] = (float)B_SZ * (-log_sigmoid(s0[0]) - log_sigmoid(-s1[0])) + s2[0];
  }
}

extern "C" void kernel_launch(void* const* d_in, const int* in_sizes, int n_in,
                              void* d_out, int out_size, void* d_ws, size_t ws_size,
                              hipStream_t stream) {
  const int*   h    = (const int*)d_in[0];
  const int*   r    = (const int*)d_in[1];
  const int*   t    = (const int*)d_in[2];
  const int*   hneg = (const int*)d_in[3];
  const int*   rneg = (const int*)d_in[4];
  const int*   tneg = (const int*)d_in[5];
  const int*   nctx = (const int*)d_in[6];
  const int*   pctx = (const int*)d_in[7];
  const float* ent  = (const float*)d_in[8];
  const float* rel  = (const float*)d_in[9];

  float* ws    = (float*)d_ws;            // 3 * NBLK floats = 12 KB
  float* o_gnp = ws;
  float* o_gnn = ws + NBLK;
  float* o_pl  = ws + 2 * NBLK;

  ptranse_main<<<NBLK, BLOCK_THREADS, 0, stream>>>(h, r, t, hneg, rneg, tneg,
                                                   nctx, pctx, ent, rel,
                                                   o_gnp, o_gnn, o_pl);
  ptranse_finish<<<1, 256, 0, stream>>>(o_gnp, o_gnn, o_pl, (float*)d_out);

  (void)in_sizes; (void)n_in; (void)out_size; (void)ws_size;
}